// Retriever_47648367182098
// MI455X (gfx1250) — compile-verified
//
#include <hip/hip_runtime.h>
#include <hip/hip_bf16.h>
#include <math.h>

typedef __attribute__((ext_vector_type(2))) float v2f;
typedef __attribute__((ext_vector_type(8))) float v8f;

#define BATCH 64
#define LSEQ 128
#define CIN 8
#define DMODEL 64
#define CDBCH 9
#define TOPK 16
#define NEG_INF (-3.402823466e38f)

// ---------------------------------------------------------------------------
// Encoder: conv branches + GELU + mean-pool + linear + LayerNorm + L2 norm
// One block per batch element. Writes normalized bx[64][64] to workspace.
// ---------------------------------------------------------------------------
__global__ void encoder_kernel(const float* __restrict__ x,
    const float* __restrict__ w1, const float* __restrict__ b1,
    const float* __restrict__ w2, const float* __restrict__ b2,
    const float* __restrict__ w3, const float* __restrict__ b3,
    const float* __restrict__ w4, const float* __restrict__ b4,
    const float* __restrict__ wf, float* __restrict__ bx)
{
    __shared__ float s_part[256];
    __shared__ float s_vec[DMODEL];
    __shared__ float s_out[DMODEL];
    __shared__ float s_stat[2];

    int b = blockIdx.x;
    int t = threadIdx.x;
    int c = t & 63;        // output channel 0..63
    int seg = t >> 6;      // 0..3 : which quarter of L
    int branch = c >> 4;   // 0..3
    int o = c & 15;        // channel within branch
    const float* xb = x + (size_t)b * LSEQ * CIN;

    float wk0[CIN], wk1[CIN], wk2[CIN], bias;
    int dil = 1;
    if (branch == 0) {
        #pragma unroll
        for (int i = 0; i < CIN; ++i) { wk0[i] = 0.f; wk1[i] = w1[o*CIN + i]; wk2[i] = 0.f; }
        bias = b1[o];
    } else {
        const float* w  = (branch == 1) ? w2 : ((branch == 2) ? w3 : w4);
        const float* bb = (branch == 1) ? b2 : ((branch == 2) ? b3 : b4);
        #pragma unroll
        for (int i = 0; i < CIN; ++i) {
            wk0[i] = w[o*CIN*3 + i*3 + 0];
            wk1[i] = w[o*CIN*3 + i*3 + 1];
            wk2[i] = w[o*CIN*3 + i*3 + 2];
        }
        bias = bb[o];
        dil = (branch == 1) ? 1 : ((branch == 2) ? 2 : 4);
    }

    float sum = 0.f;
    for (int li = 0; li < 32; ++li) {
        int l = seg * 32 + li;
        float f = bias;
        const float* xl = xb + l * CIN;
        #pragma unroll
        for (int i = 0; i < CIN; ++i) f += xl[i] * wk1[i];
        if (branch != 0) {
            int lm = l - dil, lp = l + dil;
            if (lm >= 0) {
                const float* xm = xb + lm*CIN;
                #pragma unroll
                for (int i = 0; i < CIN; ++i) f += xm[i] * wk0[i];
            }
            if (lp < LSEQ) {
                const float* xp = xb + lp*CIN;
                #pragma unroll
                for (int i = 0; i < CIN; ++i) f += xp[i] * wk2[i];
            }
        }
        // exact GELU: 0.5*x*(1+erf(x/sqrt(2)))
        float g = 0.5f * f * (1.0f + erff(f * 0.70710678118654752440f));
        sum += g;
    }
    s_part[t] = sum;
    __syncthreads();
    if (seg == 0) {
        float v = s_part[c] + s_part[64 + c] + s_part[128 + c] + s_part[192 + c];
        s_vec[c] = v * (1.0f / (float)LSEQ);       // AdaptiveAvgPool1d(1)
    }
    __syncthreads();
    if (t < DMODEL) {                              // out = vec @ wf.T
        float acc = 0.f;
        for (int j = 0; j < DMODEL; ++j) acc += s_vec[j] * wf[t*DMODEL + j];
        s_out[t] = acc;
    }
    __syncthreads();
    if (t == 0) {                                  // LayerNorm stats (biased var)
        float mu = 0.f;
        for (int d = 0; d < DMODEL; ++d) mu += s_out[d];
        mu *= (1.0f / (float)DMODEL);
        float var = 0.f;
        for (int d = 0; d < DMODEL; ++d) { float dd = s_out[d] - mu; var += dd*dd; }
        var *= (1.0f / (float)DMODEL);
        s_stat[0] = mu;
        s_stat[1] = 1.0f / sqrtf(var + 1e-5f);
    }
    __syncthreads();
    if (t < DMODEL) s_out[t] = (s_out[t] - s_stat[0]) * s_stat[1];
    __syncthreads();
    if (t == 0) {                                  // row L2 normalize
        float ss = 0.f;
        for (int d = 0; d < DMODEL; ++d) ss += s_out[d] * s_out[d];
        s_stat[0] = 1.0f / fmaxf(sqrtf(ss), 1e-12f);
    }
    __syncthreads();
    if (t < DMODEL) bx[b*DMODEL + t] = s_out[t] * s_stat[0];
}

// ---------------------------------------------------------------------------
// scores[64][N] = bx[64][64] @ ax[N][64]^T using V_WMMA_F32_16X16X4_F32.
// Register-pressure-aware layout:
//   - wave owns one 16-row M-tile (mt = wave&3): A fragments = 16 x v2f
//     (32 VGPRs), preloaded from LDS once, live across the whole tile loop.
//   - waves pair into 2 column groups (cg = wave>>2): block covers 2 column
//     tiles per iteration; single v8f accumulator per wave.
//   - B fragments streamed from global inside the K loop (L2/WGP$ absorbs
//     the 4-wave redundancy within a block).
// Fragment layouts per ISA 7.12.2 (f32):
//   A (16x4): v[0] lanes0-15 K=0 | lanes16-31 K=2 ; v[1] K=1 | K=3
//   B (4x16): symmetric (lane = N, half-wave selects K pair)
//   C/D (16x16): vgpr r: lanes0-15 -> M=r, lanes16-31 -> M=r+8, N=lane%16
// ---------------------------------------------------------------------------
__global__ void scores_wmma_kernel(const float* __restrict__ ax,
                                   const float* __restrict__ bx,
                                   float* __restrict__ scores,
                                   int N, int ntiles)
{
    __shared__ float s_bx[DMODEL * DMODEL];
    int t = threadIdx.x;
    for (int i = t; i < DMODEL * DMODEL; i += 256) s_bx[i] = bx[i];
    __syncthreads();

    int wave  = t >> 5;       // wave32
    int lane  = t & 31;
    int laneh = lane & 15;
    int hi    = lane >> 4;    // 0: K pair {0,1}   1: K pair {2,3}
    int mt    = wave & 3;     // M-tile owned by this wave
    int cg    = wave >> 2;    // column group 0/1
    int mbase = mt << 4;

    // Preload this wave's A fragments (tile-invariant, 32 VGPRs)
    v2f Af[16];
    #pragma unroll
    for (int kk = 0; kk < 16; ++kk)
        Af[kk] = *(const v2f*)(&s_bx[(mbase + laneh) * DMODEL + 4*kk + 2*hi]);

    int tileStride = gridDim.x * 2;
    for (int tile = blockIdx.x * 2 + cg; tile < ntiles; tile += tileStride) {
        int n0 = tile << 4;
        int colLoad = n0 + laneh; if (colLoad >= N) colLoad = N - 1;   // clamp (N%16==0 normally)
        const float* arow = ax + (size_t)colLoad * DMODEL + 2 * hi;

        // prefetch this lane's row for the next tile this wave will process
        int nt = tile + tileStride;
        if (nt < ntiles) {
            __builtin_prefetch(ax + (size_t)((nt << 4) + laneh) * DMODEL, 0, 1);
        }

        v8f acc = {};
        #pragma unroll
        for (int kk = 0; kk < 16; ++kk) {
            v2f b = *(const v2f*)(arow + 4 * kk);
            acc = __builtin_amdgcn_wmma_f32_16x16x4_f32(false, Af[kk], false, b,
                                                        (short)0, acc, false, false);
        }

        int col = n0 + laneh;
        if (col < N) {
            #pragma unroll
            for (int r = 0; r < 8; ++r) {
                int row = mbase + hi * 8 + r;
                scores[(size_t)row * N + col] = acc[r];
            }
        }
    }
}

// ---------------------------------------------------------------------------
// Exact top-16 per row: per-thread sorted top-16 over strided scan, then 16
// rounds of block-wide argmax over the 256 sorted-list heads. Output sorted
// descending (matches lax.top_k).
// ---------------------------------------------------------------------------
__global__ void topk_kernel(const float* __restrict__ scores, int N,
                            float* __restrict__ outScores, int* __restrict__ outIdx)
{
    __shared__ float cv[256 * TOPK];
    __shared__ int   ci[256 * TOPK];
    __shared__ float hv[256];
    __shared__ int   hidx[256];
    __shared__ float rv[256];
    __shared__ int   rt[256];

    int row = blockIdx.x;
    int t = threadIdx.x;
    const float* srow = scores + (size_t)row * N;

    float s[TOPK]; int id[TOPK];
    #pragma unroll
    for (int j = 0; j < TOPK; ++j) { s[j] = NEG_INF; id[j] = 0; }

    for (int n = t; n < N; n += 256) {
        float v = srow[n];
        if (v > s[TOPK - 1]) {
            s[TOPK - 1] = v; id[TOPK - 1] = n;
            #pragma unroll
            for (int j = TOPK - 1; j > 0; --j) {
                if (s[j] > s[j-1]) {
                    float tv = s[j]; s[j] = s[j-1]; s[j-1] = tv;
                    int   ti = id[j]; id[j] = id[j-1]; id[j-1] = ti;
                }
            }
        }
    }
    #pragma unroll
    for (int j = 0; j < TOPK; ++j) { cv[t*TOPK + j] = s[j]; ci[t*TOPK + j] = id[j]; }
    int p = 0;
    hv[t] = s[0]; hidx[t] = id[0];
    __syncthreads();

    for (int k = 0; k < TOPK; ++k) {
        rv[t] = hv[t]; rt[t] = t;
        __syncthreads();
        for (int off = 128; off > 0; off >>= 1) {
            if (t < off) {
                if (rv[t + off] > rv[t]) { rv[t] = rv[t + off]; rt[t] = rt[t + off]; }
            }
            __syncthreads();
        }
        int wt = rt[0];
        if (t == 0) {
            outScores[row*TOPK + k] = rv[0];
            outIdx[row*TOPK + k]    = hidx[wt];
        }
        __syncthreads();
        if (t == wt) {
            ++p;
            if (p < TOPK) { hv[t] = cv[t*TOPK + p]; hidx[t] = ci[t*TOPK + p]; }
            else          { hv[t] = NEG_INF; hidx[t] = 0; }
        }
        __syncthreads();
    }
}

// ---------------------------------------------------------------------------
// Gather + transpose: out[b,j,l,c] = windows[idx[b,j], c, l]
// ---------------------------------------------------------------------------
__global__ void gather_kernel(const float* __restrict__ windows,
                              const int* __restrict__ idx,
                              float* __restrict__ out)
{
    int pair = blockIdx.x;          // b*TOPK + j
    int l = threadIdx.x;            // 0..127
    int n = idx[pair];
    const float* src = windows + (size_t)n * CDBCH * LSEQ + l;
    float* dst = out + ((size_t)pair * LSEQ + l) * CDBCH;
    #pragma unroll
    for (int c = 0; c < CDBCH; ++c) dst[c] = src[c * LSEQ];
}

// ---------------------------------------------------------------------------
extern "C" void kernel_launch(void* const* d_in, const int* in_sizes, int n_in,
                              void* d_out, int out_size, void* d_ws, size_t ws_size,
                              hipStream_t stream)
{
    const float* x   = (const float*)d_in[0];
    const float* ax  = (const float*)d_in[1];
    const float* win = (const float*)d_in[2];
    const float* w1  = (const float*)d_in[3];
    const float* b1  = (const float*)d_in[4];
    const float* w2  = (const float*)d_in[5];
    const float* b2  = (const float*)d_in[6];
    const float* w3  = (const float*)d_in[7];
    const float* b3  = (const float*)d_in[8];
    const float* w4  = (const float*)d_in[9];
    const float* b4  = (const float*)d_in[10];
    const float* wf  = (const float*)d_in[11];
    int N = in_sizes[1] / DMODEL;          // 200000

    float* out    = (float*)d_out;                          // [1024 topk scores][windows]
    float* bx     = (float*)d_ws;                           // 64*64 f32
    int*   tidx   = (int*)((char*)d_ws + 16384);            // 64*16 int
    float* scores = (float*)((char*)d_ws + 32768);          // 64*N f32 (~51.2 MB)

    encoder_kernel<<<BATCH, 256, 0, stream>>>(x, w1, b1, w2, b2, w3, b3, w4, b4, wf, bx);

    int ntiles = (N + 15) >> 4;                  // 12500
    int blocks = (ntiles + 1) >> 1;              // 2 tiles per block-iteration
    scores_wmma_kernel<<<blocks, 256, 0, stream>>>(ax, bx, scores, N, ntiles);

    topk_kernel<<<BATCH, 256, 0, stream>>>(scores, N, out, tidx);

    gather_kernel<<<BATCH * TOPK, LSEQ, 0, stream>>>(win, tidx, out + BATCH * TOPK);
}